// PPOAdvatageModel_970662609487
// MI455X (gfx1250) — compile-verified
//
#include <hip/hip_runtime.h>
#include <hip/hip_bf16.h>

// ---------------------------------------------------------------------------
// PPO advantage model, MI455X (gfx1250).
//
// Roofline: dominant cost = streaming the needed rows of `hidden`
// (32 batches x 1025 rows x 2048 f32 = 268 MB) once at 23.3 TB/s (~11.5 us).
// We fuse both GEMVs (h.w_base and h.w_adapter) into a single pass using
// V_WMMA_F32_16X16X4_F32 (full fp32 matrix pipe): A = 16 rows x 4 K of
// hidden, B = 4x16 with col0 = w_base, col1 = w_adapter. Each 256-thread
// block (8 wave32s) owns a 16-row tile; each wave reduces a K=256 slice with
// 4 interleaved WMMA accumulator chains; waves combine via LDS.
// Everything downstream is O(B*R)=32K elements.
// ---------------------------------------------------------------------------

typedef float v2f __attribute__((ext_vector_type(2)));
typedef float v8f __attribute__((ext_vector_type(8)));

#define B_ 32
#define Q_ 512
#define R_ 1024
#define L_ 1536
#define D_ 2048

#define GAMMA_ 0.99f
#define LMBDA_ 0.95f
#define KLC_   0.1f
#define PEN_   (-3.0f)

// d_out layout (floats, reference tuple order)
#define OFF_VALUES 0
#define OFF_VPM    (B_ * R_)            // 32768
#define OFF_VLP    (2 * B_ * R_)        // 65536
#define OFF_SCORES (2 * B_ * R_ + B_)   // 65568
#define OFF_RLP    (2 * B_ * R_ + 2*B_) // 65600
#define OFF_KL     (2 * B_ * R_ + 3*B_) // 65632
#define OFF_KLR    (3 * B_ * R_ + 3*B_) // 98400
#define OFF_ADV    (4 * B_ * R_ + 3*B_) // 131168
#define OFF_RET    (5 * B_ * R_ + 3*B_) // 163936

// ---------------------------------------------------------------------------
// Phase A: fused dual GEMV over hidden rows l in [Q-1, L-1] via f32 WMMA.
// grid = (65 tiles of 16 rows, B), block = 256 (8 waves).
// Writes vals_raw[b][t] = (h . (w_base+w_adapter)) at l = Q-1+t   (t<1024)
//        scores_raw[b][t] = (h . w_base)            at l = Q+t    (t<1024)
// ---------------------------------------------------------------------------
__global__ __launch_bounds__(256) void gemv_wmma_kernel(
    const float* __restrict__ hidden, const float* __restrict__ w_base,
    const float* __restrict__ w_adapter, float* __restrict__ vals_raw,
    float* __restrict__ scores_raw) {
  __shared__ float s_red[8][256];
  __shared__ float s_fin[256];

  const int tile = blockIdx.x;       // 0..64 (tile 64 has 1 valid row)
  const int b    = blockIdx.y;
  const int tid  = threadIdx.x;
  const int wave = tid >> 5;
  const int lane = tid & 31;
  // WMMA f32 16x16x4 operand layout (wave32):
  //  A: lanes 0-15 -> M=0-15 with K={0,1}; lanes 16-31 -> M=0-15 with K={2,3}
  //  B: lanes 0-15 -> N=0-15 with K={0,1}; lanes 16-31 -> N=0-15 with K={2,3}
  const int m  = lane & 15;            // A row / B column
  const int kb = (lane >> 4) << 1;     // K sub-offset within chunk: 0 or 2
  const float selb = (m == 0) ? 1.0f : 0.0f;  // B col 0 = w_base
  const float sela = (m == 1) ? 1.0f : 0.0f;  // B col 1 = w_adapter

  int l  = (Q_ - 1) + tile * 16 + m;
  int lc = l > (L_ - 1) ? (L_ - 1) : l;         // clamp loads in last tile
  const float* rowp = hidden + ((size_t)b * L_ + lc) * (size_t)D_;
  const int k0 = wave * (D_ / 8);               // K slice per wave: 256

  const v8f zero8 = {0.f, 0.f, 0.f, 0.f, 0.f, 0.f, 0.f, 0.f};
  v8f acc[4] = {zero8, zero8, zero8, zero8};    // 4 chains hide WMMA latency

  for (int ch = 0; ch < 64; ch += 4) {
#pragma unroll
    for (int u = 0; u < 4; ++u) {
      const int k = k0 + (ch + u) * 4 + kb;
      v2f a  = *(const v2f*)(rowp + k);
      v2f wb = *(const v2f*)(w_base + k);
      v2f wa = *(const v2f*)(w_adapter + k);
      v2f bm;                                    // branch-free column select
      bm.x = wb.x * selb + wa.x * sela;
      bm.y = wb.y * selb + wa.y * sela;
      // D = A x B + C, full fp32 (8 args: neg_a, A, neg_b, B, c_mod, C, ra, rb)
      acc[u] = __builtin_amdgcn_wmma_f32_16x16x4_f32(
          false, a, false, bm, (short)0, acc[u], false, false);
    }
  }
  v8f accf = (acc[0] + acc[1]) + (acc[2] + acc[3]);
#pragma unroll
  for (int r = 0; r < 8; ++r) s_red[wave][lane + 32 * r] = accf[r];
  __syncthreads();

  // cross-wave reduction of the 16x16 tile (stored as [lane + 32*vgpr])
  {
    float s = 0.f;
#pragma unroll
    for (int w = 0; w < 8; ++w) s += s_red[w][tid];
    s_fin[tid] = s;
  }
  __syncthreads();

  // C/D layout: element (M,N) -> lane = N + 16*(M/8), vgpr = M%8
  if (tid < 16) {
    const int mm   = tid;
    const int half = mm >> 3;
    const int vr   = mm & 7;
    const float sb = s_fin[(16 * half + 0) + 32 * vr];  // h . w_base
    const float sa = s_fin[(16 * half + 1) + 32 * vr];  // h . w_adapter
    const int ll   = (Q_ - 1) + tile * 16 + mm;
    if (ll <= L_ - 2) vals_raw[b * R_ + (ll - (Q_ - 1))] = sb + sa;
    if (ll >= Q_ && ll <= L_ - 1) scores_raw[b * R_ + (ll - Q_)] = sb;
  }
}

// ---------------------------------------------------------------------------
// Phase B: per-batch-row scores/masks/rewards + segmented GAE suffix scan.
// grid = B, block = R_ (32 waves). Scan depth O(32) instead of O(1024).
// ---------------------------------------------------------------------------
__global__ __launch_bounds__(1024) void finalize_kernel(
    const float* __restrict__ gen, const float* __restrict__ ref,
    const unsigned char* __restrict__ padmask,
    const float* __restrict__ vals_raw, const float* __restrict__ scores_raw,
    float* __restrict__ adv_raw, float* __restrict__ stats,
    float* __restrict__ out) {
  const int b = blockIdx.x;
  const int t = threadIdx.x;
  __shared__ float s_vals[R_];
  __shared__ float s_scan[R_ + 32];   // stride-33 padded (bank-conflict free)
  __shared__ float s_S[32];
  __shared__ float s_carry[32];
  __shared__ float s_powc[33];        // (gamma*lambda)^k, serial products
  __shared__ float s_stats[3];
  __shared__ int   s_cnt;
  const float c = GAMMA_ * LMBDA_;

  if (t == 0) {
    s_cnt = 0;
    s_stats[0] = s_stats[1] = s_stats[2] = 0.f;
    float p = 1.f;
    for (int i = 0; i <= 32; ++i) { s_powc[i] = p; p *= c; }
  }
  __syncthreads();

  const int  gi  = b * R_ + t;
  const bool pad = padmask[gi] != 0;
  const int  validi = pad ? 0 : 1;

  // response_last_pos: mask is monotone (arange >= len) -> last = count(!pad)-1
  int wc = validi;
  for (int off = 16; off > 0; off >>= 1) wc += __shfl_down(wc, off, 32);
  if ((t & 31) == 0) atomicAdd(&s_cnt, wc);
  __syncthreads();

  const int  cnt   = s_cnt;
  const int  last  = cnt - 1;
  const bool penal = (cnt == R_);  // PENALISE_NO_EOS: no pad token at all
  const float score = penal ? PEN_ : scores_raw[b * R_ + last];
  const int vlp = (last > 0 && last < (R_ - 1)) ? last + 1 : last;

  const bool  vpm = (t == vlp) ? false : pad;
  const float v   = vpm ? 0.f : vals_raw[gi];
  s_vals[t] = v;
  const float kl  = gen[gi] - ref[gi];
  const float klr = -KLC_ * kl;
  const float rew = klr + ((t == vlp) ? score : 0.f);

  out[OFF_VALUES + gi] = v;
  out[OFF_VPM + gi]    = vpm ? 1.f : 0.f;
  out[OFF_KL + gi]     = kl;
  out[OFF_KLR + gi]    = klr;
  if (t == 0) {
    out[OFF_VLP + b]    = (float)vlp;
    out[OFF_SCORES + b] = score;
    out[OFF_RLP + b]    = (float)last;
  }
  __syncthreads();

  const float vnext = (t < R_ - 1) ? s_vals[t + 1] : 0.f;
  const float delta = rew + GAMMA_ * vnext - v;
  s_scan[t + (t >> 5)] = delta;      // element t -> slot 33*(t/32) + (t%32)
  __syncthreads();

  // pass 1: 32 lanes each serial-suffix-scan a 32-element segment in-place
  if (t < 32) {
    float a = 0.f;
    const int base = t * 33;
    for (int j = 31; j >= 0; --j) {
      a = s_scan[base + j] + c * a;
      s_scan[base + j] = a;
    }
    s_S[t] = a;                      // full local suffix sum of segment
  }
  __syncthreads();
  // segment carry chain: F_seg = S_seg + c^32 * F_{seg+1}; carry_in = F_{seg+1}
  if (t == 0) {
    const float c32 = s_powc[32];
    float F = 0.f;
    for (int seg = 31; seg >= 0; --seg) { s_carry[seg] = F; F = s_S[seg] + c32 * F; }
  }
  __syncthreads();
  // pass 2: adv[t] = local[t] + c^(32-j) * carry_in[seg]
  const int j = t & 31, seg = t >> 5;
  const float adv = s_scan[seg * 33 + j] + s_powc[32 - j] * s_carry[seg];
  adv_raw[gi] = adv;
  out[OFF_RET + gi] = adv + v;       // returns use un-whitened advantages

  // global whitening stats (mask = ~responses_pad_mask)
  float n1  = validi ? 1.f : 0.f;
  float sx  = validi ? adv : 0.f;
  float sxx = validi ? adv * adv : 0.f;
  for (int off = 16; off > 0; off >>= 1) {
    n1  += __shfl_down(n1, off, 32);
    sx  += __shfl_down(sx, off, 32);
    sxx += __shfl_down(sxx, off, 32);
  }
  if ((t & 31) == 0) {
    atomicAdd(&s_stats[0], n1);
    atomicAdd(&s_stats[1], sx);
    atomicAdd(&s_stats[2], sxx);
  }
  __syncthreads();
  if (t == 0) {
    atomicAdd(&stats[0], s_stats[0]);
    atomicAdd(&stats[1], s_stats[1]);
    atomicAdd(&stats[2], s_stats[2]);
  }
}

// ---------------------------------------------------------------------------
// Phase C: masked whitening. var = (S2/n - mean^2) * n/(n-1).
// ---------------------------------------------------------------------------
__global__ void whiten_kernel(const float* __restrict__ adv_raw,
                              const unsigned char* __restrict__ padmask,
                              const float* __restrict__ stats,
                              float* __restrict__ out) {
  const int i = blockIdx.x * blockDim.x + threadIdx.x;
  if (i >= B_ * R_) return;
  const float n    = stats[0];
  const float mean = stats[1] / n;
  const float var  = (stats[2] / n - mean * mean) * (n / (n - 1.f));
  const bool valid = padmask[i] == 0;
  const float w = (adv_raw[i] - mean) * rsqrtf(var + 1e-8f);
  out[OFF_ADV + i] = valid ? w : 0.f;
}

__global__ void init_stats_kernel(float* __restrict__ stats) {
  if (threadIdx.x < 3) stats[threadIdx.x] = 0.f;
}

// ---------------------------------------------------------------------------
extern "C" void kernel_launch(void* const* d_in, const int* in_sizes, int n_in,
                              void* d_out, int out_size, void* d_ws, size_t ws_size,
                              hipStream_t stream) {
  (void)in_sizes; (void)n_in; (void)out_size; (void)ws_size;
  const float* hidden        = (const float*)d_in[0];
  const float* w_base        = (const float*)d_in[1];
  const float* w_adapter     = (const float*)d_in[2];
  const float* gen_logprobs  = (const float*)d_in[3];
  const float* ref_logprobs  = (const float*)d_in[4];
  const unsigned char* pmask = (const unsigned char*)d_in[5];  // jnp bool = 1B
  // d_in[6] = queries_len (compile-time constant Q_=512, unused at runtime)

  float* out = (float*)d_out;
  float* ws  = (float*)d_ws;
  float* stats      = ws;                    // 3 floats (16-float aligned pad)
  float* vals_raw   = ws + 16;               // B*R
  float* scores_raw = ws + 16 + B_ * R_;     // B*R
  float* adv_raw    = ws + 16 + 2 * B_ * R_; // B*R

  init_stats_kernel<<<1, 32, 0, stream>>>(stats);
  gemv_wmma_kernel<<<dim3(65, B_), 256, 0, stream>>>(hidden, w_base, w_adapter,
                                                     vals_raw, scores_raw);
  finalize_kernel<<<B_, R_, 0, stream>>>(gen_logprobs, ref_logprobs, pmask,
                                         vals_raw, scores_raw, adv_raw, stats, out);
  whiten_kernel<<<(B_ * R_ + 255) / 256, 256, 0, stream>>>(adv_raw, pmask, stats, out);
}